// HeteroGNN_35184372088981
// MI455X (gfx1250) — compile-verified
//
#include <hip/hip_runtime.h>
#include <hip/hip_bf16.h>
#include <math.h>

typedef float v2f __attribute__((ext_vector_type(2)));
typedef float v8f __attribute__((ext_vector_type(8)));

#define D_HID 128
#define D_OUT 300
#define D_OUT_PAD 304
#define KPAIRS 64   // 128 / 2

// ---------------------------------------------------------------- CDNA5 async copy to LDS
// global_load_async_to_lds_b128: VDST = VGPR with LDS byte address (generic addr low 32
// bits == LDS offset per aperture rule), VADDR = 64-bit global address. Tracked by ASYNCcnt.
__device__ __forceinline__ void lds_cp16_async(void* lds_dst, const void* gsrc) {
  unsigned laddr = (unsigned)(uintptr_t)lds_dst;
  asm volatile("global_load_async_to_lds_b128 %0, %1, off"
               :: "v"(laddr), "v"(gsrc) : "memory");
}
__device__ __forceinline__ void lds_async_wait() {
  asm volatile("s_wait_asynccnt 0x0" ::: "memory");
}

// ---------------------------------------------------------------- utilities
__global__ void k_zero(float* __restrict__ p, long n) {
  long i = (long)blockIdx.x * blockDim.x + threadIdx.x;
  long stride = (long)gridDim.x * blockDim.x;
  for (; i < n; i += stride) p[i] = 0.0f;
}

// per-channel BN fold: s = gamma*rsqrt(var+eps), shift = (bl+bsk-mean)*s+beta
__global__ void k_prep_chan(const float* __restrict__ bl, const float* __restrict__ bsk,
                            const float* __restrict__ gamma, const float* __restrict__ beta,
                            const float* __restrict__ rmean, const float* __restrict__ rvar,
                            float* __restrict__ s, float* __restrict__ shift) {
  int c = threadIdx.x;
  if (c < D_HID) {
    float sc = gamma[c] * rsqrtf(rvar[c] + 1e-5f);
    s[c] = sc;
    shift[c] = (bl[c] + bsk[c] - rmean[c]) * sc + beta[c];
  }
}

// pack hidden-layer weights into WMMA B layout: p[kpair][col] = {W[2k][col], W[2k+1][col]}
__global__ void k_pack_hid(const float* __restrict__ Wl, const float* __restrict__ Wr,
                           const float* __restrict__ Wsk,
                           float* __restrict__ pWl, float* __restrict__ pWc) {
  int idx = blockIdx.x * blockDim.x + threadIdx.x;   // KPAIRS * 128
  if (idx >= KPAIRS * D_HID) return;
  int kp = idx >> 7, col = idx & 127;
  int k0 = kp * 2;
  pWl[idx * 2 + 0] = Wl[(long)k0 * D_HID + col];
  pWl[idx * 2 + 1] = Wl[(long)(k0 + 1) * D_HID + col];
  pWc[idx * 2 + 0] = Wr[(long)k0 * D_HID + col] + Wsk[(long)k0 * D_HID + col];
  pWc[idx * 2 + 1] = Wr[(long)(k0 + 1) * D_HID + col] + Wsk[(long)(k0 + 1) * D_HID + col];
}

// pack output-layer weights (300 cols -> 304, zero-padded)
__global__ void k_pack_out(const float* __restrict__ Wlo, const float* __restrict__ Wro,
                           float* __restrict__ pWlo, float* __restrict__ pWro) {
  int idx = blockIdx.x * blockDim.x + threadIdx.x;   // KPAIRS * 304
  if (idx >= KPAIRS * D_OUT_PAD) return;
  int kp = idx / D_OUT_PAD, col = idx % D_OUT_PAD;
  int k0 = kp * 2;
  float a0 = 0.f, a1 = 0.f, b0 = 0.f, b1 = 0.f;
  if (col < D_OUT) {
    a0 = Wlo[(long)k0 * D_OUT + col];
    a1 = Wlo[(long)(k0 + 1) * D_OUT + col];
    b0 = Wro[(long)k0 * D_OUT + col];
    b1 = Wro[(long)(k0 + 1) * D_OUT + col];
  }
  pWlo[idx * 2 + 0] = a0; pWlo[idx * 2 + 1] = a1;
  pWro[idx * 2 + 0] = b0; pWro[idx * 2 + 1] = b1;
}

// ---------------------------------------------------------------- graph ops
__global__ void k_degree(const int* __restrict__ ei, float* __restrict__ cnt, int E) {
  int e = blockIdx.x * blockDim.x + threadIdx.x;
  if (e < E) unsafeAtomicAdd(&cnt[ei[E + e]], 1.0f);   // dst = ei[1][e]
}

// one wave per edge; lane handles 4 consecutive feats (float4 gather + 4 f32 atomics)
__global__ void k_scatter(const int* __restrict__ ei, const float* __restrict__ feat,
                          float* __restrict__ agg, int E) {
  long t = (long)blockIdx.x * blockDim.x + threadIdx.x;
  if (t >= (long)E * 32) return;
  int e = (int)(t >> 5);
  int c = ((int)t & 31) << 2;
  int src = ei[e];
  int dst = ei[E + e];
  const float4 v = *(const float4*)(feat + (long)src * D_HID + c);
  float* a = agg + (long)dst * D_HID + c;
  unsafeAtomicAdd(a + 0, v.x);
  unsafeAtomicAdd(a + 1, v.y);
  unsafeAtomicAdd(a + 2, v.z);
  unsafeAtomicAdd(a + 3, v.w);
}

// ---------------------------------------------------------------- WMMA GEMMs
// Layer 1: h = (mean@Wl + x@Wc)*s + shift ; out_h = h ; x_emb = relu(h)
// Weights (pWl||pWc, 32768 floats = 128 KB) staged in LDS via async copy, once per block.
__global__ __launch_bounds__(256) void k_gemm1(
    const float* __restrict__ x, const float* __restrict__ agg, const float* __restrict__ cnt,
    const float* __restrict__ pW /* pWl followed by pWc */,
    const float* __restrict__ s, const float* __restrict__ shift,
    float* __restrict__ out_h, float* __restrict__ x_emb, int nTiles, int n) {
  extern __shared__ __align__(16) float lds[];
  // -------- stage 128 KB of packed weights into LDS (async, ASYNCcnt)
  {
    const float4* g = (const float4*)pW;   // 8192 float4
    float4* s4 = (float4*)lds;
    for (int i = threadIdx.x; i < 8192; i += 256)
      lds_cp16_async(s4 + i, g + i);
    lds_async_wait();
  }
  __syncthreads();

  int lane = threadIdx.x & 31;
  int wave = threadIdx.x >> 5;
  int tile = blockIdx.x * 8 + wave;
  if (tile >= nTiles) return;
  int r = lane & 15, khalf = lane >> 4;
  long rowA = (long)tile * 16 + r;
  if (rowA >= n) rowA = n - 1;                       // clamp loads for partial tile
  float invc = 1.0f / fmaxf(cnt[rowA], 1.0f);
  const float* aggRow = agg + rowA * D_HID;
  const float* xRow   = x   + rowA * D_HID;
  const float* sWl = lds;                 // 16384 floats
  const float* sWc = lds + 16384;

  v8f zero = {};
  v8f acc[8];
#pragma unroll
  for (int ct = 0; ct < 8; ++ct) acc[ct] = zero;

  // software-pipelined A operand (global), B operands from LDS
  float2 ag = *(const float2*)(aggRow + khalf * 2);
  float2 xx = *(const float2*)(xRow   + khalf * 2);
  for (int kt = 0; kt < 32; ++kt) {                  // K = 128 in steps of 4
    v2f am; am.x = ag.x * invc; am.y = ag.y * invc;
    v2f ax; ax.x = xx.x;        ax.y = xx.y;
    if (kt < 31) {
      int k0n = (kt + 1) * 4 + khalf * 2;            // prefetch next A pair
      ag = *(const float2*)(aggRow + k0n);
      xx = *(const float2*)(xRow   + k0n);
    }
    int kp = kt * 2 + khalf;
    const v2f* bL = (const v2f*)(sWl + kp * (2 * D_HID));
    const v2f* bC = (const v2f*)(sWc + kp * (2 * D_HID));
    v2f bw[8], bc[8];
#pragma unroll
    for (int ct = 0; ct < 8; ++ct) {                 // batched ds loads
      bw[ct] = bL[ct * 16 + r];
      bc[ct] = bC[ct * 16 + r];
    }
#pragma unroll
    for (int ct = 0; ct < 8; ++ct) {
      acc[ct] = __builtin_amdgcn_wmma_f32_16x16x4_f32(false, am, false, bw[ct],
                                                      (short)0, acc[ct], false, false);
      acc[ct] = __builtin_amdgcn_wmma_f32_16x16x4_f32(false, ax, false, bc[ct],
                                                      (short)0, acc[ct], false, false);
    }
  }
  // epilogue: D layout VGPR i -> M = i + 8*khalf, N = lane&15
#pragma unroll
  for (int ct = 0; ct < 8; ++ct) {
#pragma unroll
    for (int i = 0; i < 8; ++i) {
      long row = (long)tile * 16 + i + khalf * 8;
      int col = ct * 16 + r;
      if (row < n) {
        float h = acc[ct][i] * s[col] + shift[col];
        out_h[row * D_HID + col] = h;
        x_emb[row * D_HID + col] = fmaxf(h, 0.0f);
      }
    }
  }
}

// Layer 2: logits = mean2@Wlo + x_emb@Wro + blo. Column-slice of packed weights staged
// in LDS (NCT*16 cols -> NCT*8 KB per matrix), col tiles templated so acc stays in VGPRs.
template <int NCT>
__global__ __launch_bounds__(256) void k_gemm2(
    const float* __restrict__ xe, const float* __restrict__ agg, const float* __restrict__ cnt,
    const float* __restrict__ pWlo, const float* __restrict__ pWro,
    const float* __restrict__ blo, float* __restrict__ logits, int nTiles, int n, int ctBase) {
  extern __shared__ __align__(16) float lds[];
  float* sWlo = lds;                         // KPAIRS * NCT*32 floats
  float* sWro = lds + KPAIRS * (NCT * 32);
  // -------- stage the column slice [ctBase*16, ctBase*16 + NCT*16) of both matrices
  {
    const int PERM = KPAIRS * NCT * 8;       // float4 per matrix
    for (int i = threadIdx.x; i < PERM; i += 256) {
      int kp = i / (NCT * 8), q = i % (NCT * 8);
      lds_cp16_async((float4*)(sWlo + kp * (NCT * 32)) + q,
                     (const float4*)(pWlo + (long)kp * (2 * D_OUT_PAD) + ctBase * 32) + q);
      lds_cp16_async((float4*)(sWro + kp * (NCT * 32)) + q,
                     (const float4*)(pWro + (long)kp * (2 * D_OUT_PAD) + ctBase * 32) + q);
    }
    lds_async_wait();
  }
  __syncthreads();

  int lane = threadIdx.x & 31;
  int wave = threadIdx.x >> 5;
  int tile = blockIdx.x * 8 + wave;
  if (tile >= nTiles) return;
  int r = lane & 15, khalf = lane >> 4;
  long rowA = (long)tile * 16 + r;
  if (rowA >= n) rowA = n - 1;
  float invc = 1.0f / fmaxf(cnt[rowA], 1.0f);
  const float* aggRow = agg + rowA * D_HID;
  const float* xRow   = xe  + rowA * D_HID;

  v8f zero = {};
  v8f acc[NCT];
#pragma unroll
  for (int j = 0; j < NCT; ++j) acc[j] = zero;

  float2 ag = *(const float2*)(aggRow + khalf * 2);
  float2 xx = *(const float2*)(xRow   + khalf * 2);
  for (int kt = 0; kt < 32; ++kt) {
    v2f am; am.x = ag.x * invc; am.y = ag.y * invc;
    v2f ax; ax.x = xx.x;        ax.y = xx.y;
    if (kt < 31) {
      int k0n = (kt + 1) * 4 + khalf * 2;
      ag = *(const float2*)(aggRow + k0n);
      xx = *(const float2*)(xRow   + k0n);
    }
    int kp = kt * 2 + khalf;
    const v2f* bL = (const v2f*)(sWlo + kp * (NCT * 32));
    const v2f* bR = (const v2f*)(sWro + kp * (NCT * 32));
    v2f b1[NCT], b2[NCT];
#pragma unroll
    for (int j = 0; j < NCT; ++j) {
      b1[j] = bL[j * 16 + r];
      b2[j] = bR[j * 16 + r];
    }
#pragma unroll
    for (int j = 0; j < NCT; ++j) {
      acc[j] = __builtin_amdgcn_wmma_f32_16x16x4_f32(false, am, false, b1[j],
                                                     (short)0, acc[j], false, false);
      acc[j] = __builtin_amdgcn_wmma_f32_16x16x4_f32(false, ax, false, b2[j],
                                                     (short)0, acc[j], false, false);
    }
  }
#pragma unroll
  for (int j = 0; j < NCT; ++j) {
#pragma unroll
    for (int i = 0; i < 8; ++i) {
      long row = (long)tile * 16 + i + khalf * 8;
      int col = (ctBase + j) * 16 + r;
      if (row < n && col < D_OUT)
        logits[row * D_OUT + col] = acc[j][i] + blo[col];
    }
  }
}

// in-place log_softmax over 300 cols; one wave32 per row, shfl_xor reductions
__global__ __launch_bounds__(256) void k_lsm(float* __restrict__ logits, int n) {
  int lane = threadIdx.x & 31;
  int wave = threadIdx.x >> 5;
  long row = (long)blockIdx.x * 8 + wave;
  if (row >= n) return;
  float* p = logits + row * D_OUT;
  float mx = -__builtin_inff();
  for (int c = lane; c < D_OUT; c += 32) mx = fmaxf(mx, p[c]);
#pragma unroll
  for (int o = 16; o > 0; o >>= 1) mx = fmaxf(mx, __shfl_xor(mx, o, 32));
  float sum = 0.0f;
  for (int c = lane; c < D_OUT; c += 32) sum += expf(p[c] - mx);
#pragma unroll
  for (int o = 16; o > 0; o >>= 1) sum += __shfl_xor(sum, o, 32);
  float lse = mx + logf(sum);
  for (int c = lane; c < D_OUT; c += 32) p[c] = p[c] - lse;
}

// ---------------------------------------------------------------- host entry
extern "C" void kernel_launch(void* const* d_in, const int* in_sizes, int n_in,
                              void* d_out, int out_size, void* d_ws, size_t ws_size,
                              hipStream_t stream) {
  const float* x     = (const float*)d_in[0];
  const int*   ei    = (const int*)  d_in[1];
  const float* Wl    = (const float*)d_in[2];
  const float* bl    = (const float*)d_in[3];
  const float* Wr    = (const float*)d_in[4];
  const float* Wsk   = (const float*)d_in[5];
  const float* bsk   = (const float*)d_in[6];
  const float* gamma = (const float*)d_in[7];
  const float* beta  = (const float*)d_in[8];
  const float* rmean = (const float*)d_in[9];
  const float* rvar  = (const float*)d_in[10];
  const float* Wlo   = (const float*)d_in[11];
  const float* blo   = (const float*)d_in[12];
  const float* Wro   = (const float*)d_in[13];

  const int N = in_sizes[0] / D_HID;
  const int E = in_sizes[1] / 2;
  const int nTiles = (N + 15) / 16;

  // workspace layout (floats)
  float* w      = (float*)d_ws;
  float* cnt    = w;                         // [N]
  float* agg    = cnt + N;                   // [N*128]  (layer1 agg, reused for layer2)
  float* x_emb  = agg + (long)N * D_HID;     // [N*128]
  float* sbuf   = x_emb + (long)N * D_HID;   // [128]
  float* shift  = sbuf + D_HID;              // [128]
  float* pWl    = shift + D_HID;             // [64*128*2]  (pWc contiguous after)
  float* pWc    = pWl + 2 * KPAIRS * D_HID;  // [64*128*2]
  float* pWlo   = pWc + 2 * KPAIRS * D_HID;  // [64*304*2]
  float* pWro   = pWlo + 2 * KPAIRS * D_OUT_PAD;

  float* out_h   = (float*)d_out;            // [N*128]
  float* logits  = out_h + (long)N * D_HID;  // [N*300]

  // 1) zero cnt + agg (contiguous)
  k_zero<<<2048, 256, 0, stream>>>(cnt, (long)N + (long)N * D_HID);
  // 2) fold BN / biases, pack weights
  k_prep_chan<<<1, D_HID, 0, stream>>>(bl, bsk, gamma, beta, rmean, rvar, sbuf, shift);
  k_pack_hid<<<(KPAIRS * D_HID + 255) / 256, 256, 0, stream>>>(Wl, Wr, Wsk, pWl, pWc);
  k_pack_out<<<(KPAIRS * D_OUT_PAD + 255) / 256, 256, 0, stream>>>(Wlo, Wro, pWlo, pWro);
  // 3) degree + layer-1 scatter
  k_degree<<<(E + 255) / 256, 256, 0, stream>>>(ei, cnt, E);
  k_scatter<<<(int)(((long)E * 32 + 255) / 256), 256, 0, stream>>>(ei, x, agg, E);
  // 4) layer-1 WMMA GEMM (+BN fold, ReLU); 128 KB dynamic LDS for weights
  k_gemm1<<<(nTiles + 7) / 8, 256, 128 * 1024, stream>>>(x, agg, cnt, pWl, sbuf, shift,
                                                         out_h, x_emb, nTiles, N);
  // 5) re-zero agg, layer-2 scatter of x_emb
  k_zero<<<2048, 256, 0, stream>>>(agg, (long)N * D_HID);
  k_scatter<<<(int)(((long)E * 32 + 255) / 256), 256, 0, stream>>>(ei, x_emb, agg, E);
  // 6) layer-2 WMMA GEMM, split over 19 column tiles (10 + 9); LDS = NCT*16 KB
  k_gemm2<10><<<(nTiles + 7) / 8, 256, 10 * 16384, stream>>>(x_emb, agg, cnt, pWlo, pWro,
                                                             blo, logits, nTiles, N, 0);
  k_gemm2<9><<<(nTiles + 7) / 8, 256, 9 * 16384, stream>>>(x_emb, agg, cnt, pWlo, pWro,
                                                           blo, logits, nTiles, N, 10);
  // 7) log_softmax in place
  k_lsm<<<(N + 7) / 8, 256, 0, stream>>>(logits, N);
}